// Linear_block_33002528703049
// MI455X (gfx1250) — compile-verified
//
#include <hip/hip_runtime.h>

// out[b][o] = sum_k x[b][k] * W[o][k] + bias[o]
// x: (1048576, 128) f32, W: (128, 128) f32 row-major (out, in), bias: (128,1) f32.
//
// MI455X (gfx1250, wave32) strategy:
//  - HBM-bound (x read 512MB + out write 512MB @ 23.3 TB/s ~ 46us floor).
//  - V_WMMA_F32_16X16X4_F32, full f32 precision.
//  - W staged once per block into LDS in *fragment order*:
//      frag(kk, t) for lane L at dword (kk*8 + t)*64 + L*2
//    -> every ds_load_b64 is base(lane*2) + 16-bit immediate offset,
//       contiguous 256B per access, banks {2L,2L+1}: conflict-free,
//       pairs into ds_load_2addr_b64.
//  - M=32 rows per wave (two 16-row tiles) so each B fragment feeds 2 WMMAs:
//    halves LDS read traffic per output (LDS was nearing co-limiter status).
//  - A (16x4 f32) layout: lane L holds row M=L%16; K pair picked by L/16.
//  - C/D (16x16 f32): VGPR r, lane L -> M = r + 8*(L/16), N = L%16.
//  - out stores non-temporal (write-once).

typedef __attribute__((ext_vector_type(2))) float v2f;
typedef __attribute__((ext_vector_type(4))) float v4f;
typedef __attribute__((ext_vector_type(8))) float v8f;

#define BATCH   1048576
#define IN_SZ   128
#define OUT_SZ  128

__global__ __launch_bounds__(256) void batched_vmm_wmma_f32(
    const float* __restrict__ x,
    const float* __restrict__ W,
    const float* __restrict__ bias,
    float* __restrict__ out)
{
    // W in fragment order: 32 chunks x 8 tiles x 32 lanes x v2f = 64 KB.
    __shared__ float wl[OUT_SZ * IN_SZ];

    const int tid  = threadIdx.x;
    const int lane = tid & 31;
    const int wave = tid >> 5;
    const int l16  = lane & 15;      // M (A) / N (B,C,D) within a tile
    const int half = lane >> 4;      // K-pair select (A/B) / M+8 (C,D)

    // ---- Cooperative fill: coalesced b128 reads, permuted ds_store_b64 writes.
#pragma unroll
    for (int j = 0; j < 16; ++j) {
        int c4    = tid + j * 256;
        int row   = c4 >> 5;                      // 32 chunks per 128-float row
        int chunk = c4 & 31;
        v4f v = *(const v4f*)(W + (size_t)c4 * 4);   // coalesced b128
        int d0 = ((chunk * 8 + (row >> 4)) * 32 + (row & 15)) * 2;
        *(v2f*)(wl + d0)      = (v2f){v.x, v.y};     // half=0 lane
        *(v2f*)(wl + d0 + 32) = (v2f){v.z, v.w};     // half=1 lane (+16 lanes)
    }
    __syncthreads();

    const int rowBase = blockIdx.x * 256 + wave * 32;   // 32 batch rows per wave
    const float* xrow0 = x + (size_t)(rowBase + l16) * IN_SZ + 2 * half;
    const float* xrow1 = xrow0 + 16 * IN_SZ;
    const float* wlane = wl + lane * 2;                 // single DS base VGPR

    // Accumulators: 2 M-tiles x 8 N-tiles; bias folded in (N = t*16 + l16).
    v8f acc0[8], acc1[8];
#pragma unroll
    for (int t = 0; t < 8; ++t) {
        float bb = bias[t * 16 + l16];
        acc0[t] = (v8f){bb, bb, bb, bb, bb, bb, bb, bb};
        acc1[t] = acc0[t];
    }

    // ---- K loop, fully unrolled: 32 chunks of K=4; B frags shared by 2 M-tiles.
#pragma unroll
    for (int kk = 0; kk < 32; ++kk) {
        v2f a0 = *(const v2f*)(xrow0 + kk * 4);          // HBM stream, tile M0
        v2f a1 = *(const v2f*)(xrow1 + kk * 4);          // HBM stream, tile M1
#pragma unroll
        for (int t = 0; t < 8; ++t) {
            v2f bfr = *(const v2f*)(wlane + (kk * 8 + t) * 64);
            acc0[t] = __builtin_amdgcn_wmma_f32_16x16x4_f32(
                false, a0, false, bfr, (short)0, acc0[t], false, false);
            acc1[t] = __builtin_amdgcn_wmma_f32_16x16x4_f32(
                false, a1, false, bfr, (short)0, acc1[t], false, false);
        }
    }

    // ---- Store: VGPR r of acc{m}[t] -> row rowBase + m*16 + r + 8*half,
    //      col t*16 + l16. r-outer/t-inner fills contiguous 512B row segments.
#pragma unroll
    for (int r = 0; r < 8; ++r) {
        float* orow0 = out + (size_t)(rowBase + r + 8 * half) * OUT_SZ + l16;
        float* orow1 = orow0 + 16 * OUT_SZ;
#pragma unroll
        for (int t = 0; t < 8; ++t) {
            __builtin_nontemporal_store(acc0[t][r], orow0 + t * 16);
            __builtin_nontemporal_store(acc1[t][r], orow1 + t * 16);
        }
    }
}

extern "C" void kernel_launch(void* const* d_in, const int* in_sizes, int n_in,
                              void* d_out, int out_size, void* d_ws, size_t ws_size,
                              hipStream_t stream) {
    (void)in_sizes; (void)n_in; (void)d_ws; (void)ws_size; (void)out_size;
    const float* x    = (const float*)d_in[0];
    const float* W    = (const float*)d_in[1];
    const float* bias = (const float*)d_in[2];
    float* out        = (float*)d_out;

    // 8 waves/block * 32 rows/wave = 256 batch rows per block.
    dim3 grid(BATCH / 256);
    dim3 block(256);
    batched_vmm_wmma_f32<<<grid, block, 0, stream>>>(x, W, bias, out);
}